// ConditioningEncoder_34583076668089
// MI455X (gfx1250) — compile-verified
//
#include <hip/hip_runtime.h>
#include <hip/hip_bf16.h>

// Shapes from the reference
#define KB 16
#define NN 512
#define HH 64
#define ZZ 128
#define TT 8

typedef __attribute__((ext_vector_type(2))) float v2f;
typedef __attribute__((ext_vector_type(4))) float v4f;
typedef __attribute__((ext_vector_type(8))) float v8f;

// D(16x16,f32) = A(16x4,f32) * B(4x16,f32) + C   -- full-precision CDNA5 WMMA
static __device__ __forceinline__ v8f wmma4(v2f a, v2f b, v8f c) {
    return __builtin_amdgcn_wmma_f32_16x16x4_f32(
        /*neg_a=*/false, a, /*neg_b=*/false, b,
        /*c_mod=*/(short)0, c, /*reuse_a=*/false, /*reuse_b=*/false);
}

// CDNA5 async global->LDS DMA (16B per lane), tracked with ASYNCcnt.
// VGLOBAL GVS form: vdst = LDS byte address, vaddr = 32-bit byte offset,
// saddr = 64-bit SGPR base.   (CDNA5 ISA)
static __device__ __forceinline__ void async_copy16(unsigned lds_byte_addr,
                                                    unsigned long long gbase,
                                                    unsigned gbyte_off) {
    asm volatile("global_load_async_to_lds_b128 %0, %1, %2"
                 :: "v"(lds_byte_addr), "v"(gbyte_off), "s"(gbase)
                 : "memory");
}
static __device__ __forceinline__ void async_wait0() {
    asm volatile("s_wait_asynccnt 0x0" ::: "memory");
}

// ---------------------------------------------------------------------------
// Kernel 1: per-(k,j) edge scalars.
//   adjS[i,j] = adj[i,j] + (i==j)
//   SA   = sum_i adjS*A      Sadj = sum_i adjS*adj
//   Sdeg = sum_i adjS        Spred = adjS[pi_final[j], j]
// A is single-use: stream it non-temporally so the 192 MB L2 keeps adj
// resident for the two mp-layer passes.
// ---------------------------------------------------------------------------
__global__ __launch_bounds__(512) void scalars_kernel(
    const float* __restrict__ A, const float* __restrict__ adj,
    const int* __restrict__ pi,
    float* __restrict__ SA, float* __restrict__ Sadj,
    float* __restrict__ Spred, float* __restrict__ Sdeg,
    float* __restrict__ invdeg)
{
    const int k = blockIdx.x;
    const int j = threadIdx.x;
    const float* Ak   = A   + (size_t)k * NN * NN;
    const float* adjk = adj + (size_t)k * NN * NN;
    float sA = 0.f, sAd = 0.f, sD = 0.f;
    for (int i = 0; i < NN; ++i) {
        float a  = __builtin_nontemporal_load(Ak + i * NN + j);  // one-shot stream
        float ad = adjk[i * NN + j];                              // L2-resident
        float as = ad + ((i == j) ? 1.f : 0.f);
        sA  += as * a;
        sAd += as * ad;
        sD  += as;
    }
    int p = pi[(TT - 1) * KB * NN + k * NN + j];
    float sp = adjk[p * NN + j] + ((p == j) ? 1.f : 0.f);
    int o = k * NN + j;
    SA[o] = sA; Sadj[o] = sAd; Spred[o] = sp; Sdeg[o] = sD;
    invdeg[o] = 1.f / fmaxf(sD, 1.f);
}

// ---------------------------------------------------------------------------
// Kernel 2: initial node features  nf[k,j,h] = [s,d0,df,df-d0] @ Wn + bn
// ---------------------------------------------------------------------------
__global__ __launch_bounds__(256) void initnf_kernel(
    const float* __restrict__ s, const float* __restrict__ dh,
    const float* __restrict__ Wn, const float* __restrict__ bn,
    float* __restrict__ nf)
{
    int idx = blockIdx.x * 256 + threadIdx.x;   // k*N*H flat
    int h = idx & (HH - 1);
    int kn = idx >> 6;                          // k*N + j
    float sv = s[kn];
    float d0 = dh[kn];
    float df = dh[(TT - 1) * KB * NN + kn];
    float dd = df - d0;
    nf[idx] = sv * Wn[h] + d0 * Wn[HH + h] + df * Wn[2 * HH + h]
            + dd * Wn[3 * HH + h] + bn[h];
}

// ---------------------------------------------------------------------------
// Kernel 3: one message-passing layer, fully WMMA.
// grid = (K, 4 slabs of 128 nodes), block = 256 (8 waves, 16 nodes/wave).
// nf (128 KB) and Wmp (32 KB) DMA'd into LDS via the async path.
// ---------------------------------------------------------------------------
__global__ __launch_bounds__(256) void mp_layer_kernel(
    const float* __restrict__ adj,
    const float* __restrict__ nf_in, float* __restrict__ nf_out,
    const float* __restrict__ SA, const float* __restrict__ Sadj,
    const float* __restrict__ Spred, const float* __restrict__ Sdeg,
    const float* __restrict__ invdeg,
    const float* __restrict__ We, const float* __restrict__ be,
    const float* __restrict__ Wmp, const float* __restrict__ bmp)
{
    __shared__ float sNF[NN * HH];        // 128 KB: all node features of this k
    __shared__ float sMsg[8 * 16 * HH];   //  32 KB: per-wave msg tiles
    __shared__ float sWmp[2 * HH * HH];   //  32 KB: MLP weights
    __shared__ float sSA[128], sSadj[128], sSpred[128], sSdeg[128], sInv[128];
    __shared__ float sWe[3 * HH], sbe[HH], sbmp[HH];

    const int k    = blockIdx.x;
    const int slab = blockIdx.y;
    const int tid  = threadIdx.x;
    const int wave = tid >> 5;
    const int lane = tid & 31;
    const int j0   = slab * 128 + wave * 16;   // first node of this wave (within k)

    // ---- stage LDS: async DMA global->LDS, no VGPR round-trip -------------
    {
        const unsigned ldsNF = (unsigned)(size_t)&sNF[0];
        const unsigned long long gnf =
            (unsigned long long)(size_t)(nf_in + (size_t)k * NN * HH);
        for (int r = tid * 4; r < NN * HH; r += 256 * 4)          // 32 x b128/lane-set
            async_copy16(ldsNF + (unsigned)(r * 4), gnf, (unsigned)(r * 4));
        const unsigned ldsW = (unsigned)(size_t)&sWmp[0];
        const unsigned long long gw = (unsigned long long)(size_t)Wmp;
        for (int r = tid * 4; r < 2 * HH * HH; r += 256 * 4)      //  8 x b128/lane-set
            async_copy16(ldsW + (unsigned)(r * 4), gw, (unsigned)(r * 4));
    }
    if (tid < 128) {
        int jg = k * NN + slab * 128 + tid;
        sSA[tid] = SA[jg]; sSadj[tid] = Sadj[jg]; sSpred[tid] = Spred[jg];
        sSdeg[tid] = Sdeg[jg]; sInv[tid] = invdeg[jg];
    }
    if (tid < HH) {
        sWe[tid] = We[tid]; sWe[HH + tid] = We[HH + tid];
        sWe[2 * HH + tid] = We[2 * HH + tid];
        sbe[tid] = be[tid]; sbmp[tid] = bmp[tid];
    }
    async_wait0();        // our async copies landed in LDS
    __syncthreads();      // everyone else's too

    // WMMA fragment coordinates (ISA 7.12.2, 32-bit layouts)
    const int m   = lane & 15;          // M (and N) index within 16-tile
    const int kk  = (lane >> 4) << 1;   // K base held by this lane: 0 or 2
    const int rob = (lane >> 4) << 3;   // C/D row offset: 0 or 8

    // ---- aggregation: out[j,h] = sum_i adjS[i,j] * nf[i,h] ---------------
    v8f acc[4] = {v8f{}, v8f{}, v8f{}, v8f{}};
    const float* adjk = adj + (size_t)k * NN * NN;
    const int j = j0 + m;
    for (int ib = 0; ib < NN; ib += 4) {
        const int i0 = ib + kk, i1 = i0 + 1;
        // hide the 2KB-strided row fetches ~16 rows ahead (wraps harmlessly)
        __builtin_prefetch(adjk + ((ib + 16) & (NN - 1)) * NN + j, 0, 1);
        v2f a;
        a.x = adjk[i0 * NN + j] + ((i0 == j) ? 1.f : 0.f);
        a.y = adjk[i1 * NN + j] + ((i1 == j) ? 1.f : 0.f);
        #pragma unroll
        for (int t = 0; t < 4; ++t) {
            v2f b;
            b.x = sNF[i0 * HH + t * 16 + m];
            b.y = sNF[i1 * HH + t * 16 + m];
            acc[t] = wmma4(a, b, acc[t]);
        }
    }

    // ---- rank-1 edge corrections + degree division -> msg tile in LDS ----
    float* msgW = sMsg + wave * 16 * HH;
    #pragma unroll
    for (int t = 0; t < 4; ++t) {
        const int h = t * 16 + m;
        #pragma unroll
        for (int v = 0; v < 8; ++v) {
            const int rl = v + rob;            // row in 16-tile
            const int jl = wave * 16 + rl;     // slab-local node
            float val = (acc[t][v]
                         + sSA[jl]   * sWe[h]
                         + sSadj[jl] * sWe[HH + h]
                         + sSpred[jl]* sWe[2 * HH + h]
                         + sSdeg[jl] * sbe[h]) * sInv[jl];
            msgW[rl * HH + h] = val;
        }
    }

    // ---- MLP: relu(concat(nf, msg) @ Wmp + bmp), K' = 128 -----------------
    v8f d[4] = {v8f{}, v8f{}, v8f{}, v8f{}};
    for (int kp = 0; kp < 2 * HH; kp += 4) {
        const int r0 = kp + kk, r1 = r0 + 1;   // never straddles the 64-boundary
        v2f a;
        a.x = (r0 < HH) ? sNF[(j0 + m) * HH + r0] : msgW[m * HH + (r0 - HH)];
        a.y = (r1 < HH) ? sNF[(j0 + m) * HH + r1] : msgW[m * HH + (r1 - HH)];
        #pragma unroll
        for (int t = 0; t < 4; ++t) {
            v2f b;
            b.x = sWmp[r0 * HH + t * 16 + m];
            b.y = sWmp[r1 * HH + t * 16 + m];
            d[t] = wmma4(a, b, d[t]);
        }
    }

    // ---- epilogue: bias + relu, write new node features -------------------
    float* outk = nf_out + (size_t)k * NN * HH;
    #pragma unroll
    for (int t = 0; t < 4; ++t) {
        const int h = t * 16 + m;
        #pragma unroll
        for (int v = 0; v < 8; ++v) {
            const int jl = j0 + v + rob;
            outk[jl * HH + h] = fmaxf(d[t][v] + sbmp[h], 0.f);
        }
    }
}

// ---------------------------------------------------------------------------
// Kernel 4: per-k mean pool over nodes, then @ Wz + bz  -> e[k, Z]
// ---------------------------------------------------------------------------
__global__ __launch_bounds__(128) void pool_kernel(
    const float* __restrict__ nf, const float* __restrict__ Wz,
    const float* __restrict__ bz, float* __restrict__ e)
{
    __shared__ float part[128];
    __shared__ float pooled[HH];
    const int k = blockIdx.x;
    const int t = threadIdx.x;
    const int h = t & (HH - 1);
    const int half = t >> 6;
    const float* nfk = nf + (size_t)k * NN * HH;
    float sum = 0.f;
    for (int j2 = half * 256; j2 < half * 256 + 256; ++j2)
        sum += nfk[j2 * HH + h];
    part[t] = sum;
    __syncthreads();
    if (t < HH) pooled[t] = (part[t] + part[HH + t]) * (1.f / (float)NN);
    __syncthreads();
    float ez = bz[t];
    for (int h2 = 0; h2 < HH; ++h2) ez += pooled[h2] * Wz[h2 * ZZ + t];
    e[k * ZZ + t] = ez;
}

// Kernel 5: z = mean over k of e[k, :]
__global__ __launch_bounds__(128) void final_kernel(
    const float* __restrict__ e, float* __restrict__ out)
{
    const int t = threadIdx.x;
    float s = 0.f;
    for (int k = 0; k < KB; ++k) s += e[k * ZZ + t];
    out[t] = s * (1.f / (float)KB);
}

// ---------------------------------------------------------------------------
extern "C" void kernel_launch(void* const* d_in, const int* in_sizes, int n_in,
                              void* d_out, int out_size, void* d_ws, size_t ws_size,
                              hipStream_t stream) {
    const float* s    = (const float*)d_in[0];
    const float* A    = (const float*)d_in[1];
    const float* adj  = (const float*)d_in[2];
    const float* dh   = (const float*)d_in[3];
    const int*   pi   = (const int*)  d_in[4];
    const float* Wn   = (const float*)d_in[5];
    const float* bn   = (const float*)d_in[6];
    const float* We   = (const float*)d_in[7];
    const float* be   = (const float*)d_in[8];
    const float* Wmp0 = (const float*)d_in[9];
    const float* bmp0 = (const float*)d_in[10];
    const float* Wmp1 = (const float*)d_in[11];
    const float* bmp1 = (const float*)d_in[12];
    const float* Wz   = (const float*)d_in[13];
    const float* bz   = (const float*)d_in[14];

    float* ws = (float*)d_ws;
    float* SA     = ws;                 // K*N
    float* Sadj   = SA     + KB * NN;
    float* Spred  = Sadj   + KB * NN;
    float* Sdeg   = Spred  + KB * NN;
    float* invdeg = Sdeg   + KB * NN;
    float* nfA    = invdeg + KB * NN;   // K*N*H
    float* nfB    = nfA + KB * NN * HH;
    float* e      = nfB + KB * NN * HH; // K*Z

    scalars_kernel<<<KB, 512, 0, stream>>>(A, adj, pi, SA, Sadj, Spred, Sdeg, invdeg);
    initnf_kernel<<<KB * NN * HH / 256, 256, 0, stream>>>(s, dh, Wn, bn, nfA);
    mp_layer_kernel<<<dim3(KB, 4), 256, 0, stream>>>(adj, nfA, nfB, SA, Sadj, Spred,
                                                     Sdeg, invdeg, We, be, Wmp0, bmp0);
    mp_layer_kernel<<<dim3(KB, 4), 256, 0, stream>>>(adj, nfB, nfA, SA, Sadj, Spred,
                                                     Sdeg, invdeg, We, be, Wmp1, bmp1);
    pool_kernel<<<KB, 128, 0, stream>>>(nfA, Wz, bz, e);
    final_kernel<<<1, 128, 0, stream>>>(e, (float*)d_out);
}